// UniformLossGeoA3_46557445488922
// MI455X (gfx1250) — compile-verified
//
#include <hip/hip_runtime.h>

// ---------------------------------------------------------------------------
// UniformLossGeoA3 for MI455X (gfx1250, wave32)
//   B=4, N=4096, npoint = N*0.05 = 204
//   percentages p = {0.016,0.024,0.032,0.040,0.048} (already *4)
//   nsample = int(N*p) = {65, 98, 131, 163, 196}
// Pipeline:
//   k1: FPS (1 block per batch, sequential 204-step argmax)
//   k2: per (perc, batch, center): ball-query compaction + WMMA Gram ->
//       per-row nearest-neighbor distance -> partial loss
//   k3: deterministic fixed-order reduction into d_out[B]
// ---------------------------------------------------------------------------

typedef __attribute__((ext_vector_type(2))) float v2f;
typedef __attribute__((ext_vector_type(8))) float v8f;

#define N_PTS      4096
#define NPOINT     204
#define NPERC      5
#define BQ_THREADS 256
#define PPT        (N_PTS / BQ_THREADS)   // 16 points per thread
#define NS16_MAX   208                    // 196 rounded up to 16
#define PI_F       3.14159265358979323846f

__constant__ float c_p[NPERC]  = {0.016f, 0.024f, 0.032f, 0.040f, 0.048f};
__constant__ int   c_ns[NPERC] = {65, 98, 131, 163, 196};

// ---------------------------------------------------------------------------
// Kernel 1: farthest point sampling (pointnet2 style, start index 0)
// ---------------------------------------------------------------------------
__global__ __launch_bounds__(BQ_THREADS)
void fps_kernel(const float* __restrict__ pts, float4* __restrict__ subpts)
{
    const int b   = blockIdx.x;
    const int tid = threadIdx.x;
    const float* bp = pts + (size_t)b * N_PTS * 3;

    float px[PPT], py[PPT], pz[PPT], dist[PPT];
#pragma unroll
    for (int k = 0; k < PPT; ++k) {
        const int p = tid * PPT + k;
        px[k] = bp[p * 3 + 0];
        py[k] = bp[p * 3 + 1];
        pz[k] = bp[p * 3 + 2];
        dist[k] = 1.0e10f;
    }

    __shared__ int   s_far;
    __shared__ float s_cx, s_cy, s_cz;
    __shared__ float s_v[BQ_THREADS];
    __shared__ int   s_i[BQ_THREADS];
    if (tid == 0) s_far = 0;
    __syncthreads();

    for (int it = 0; it < NPOINT; ++it) {
        if (tid == 0) {
            const int f = s_far;
            const float cx = bp[f * 3 + 0];
            const float cy = bp[f * 3 + 1];
            const float cz = bp[f * 3 + 2];
            s_cx = cx; s_cy = cy; s_cz = cz;
            subpts[b * NPOINT + it] = make_float4(cx, cy, cz, 0.0f);
        }
        __syncthreads();
        const float cx = s_cx, cy = s_cy, cz = s_cz;

        float bv = -1.0f;
        int   bi = 0;
#pragma unroll
        for (int k = 0; k < PPT; ++k) {
            const float dx = px[k] - cx, dy = py[k] - cy, dz = pz[k] - cz;
            const float d  = dx * dx + dy * dy + dz * dz;
            const float nd = fminf(dist[k], d);
            dist[k] = nd;
            if (nd > bv) { bv = nd; bi = tid * PPT + k; }   // strict > keeps first index
        }
        s_v[tid] = bv;
        s_i[tid] = bi;
        __syncthreads();
        for (int off = BQ_THREADS / 2; off > 0; off >>= 1) {
            if (tid < off) {
                const float v2v = s_v[tid + off];
                const int   i2  = s_i[tid + off];
                const float v1  = s_v[tid];
                const int   i1  = s_i[tid];
                if (v2v > v1 || (v2v == v1 && i2 < i1)) { s_v[tid] = v2v; s_i[tid] = i2; }
            }
            __syncthreads();
        }
        if (tid == 0) s_far = s_i[0];
        __syncthreads();
    }
}

// ---------------------------------------------------------------------------
// Kernel 2: one block per (percentage, batch, center).
//   Phase A: ball query -> ordered compaction of first nsample in-radius
//            points into LDS, pad with first point / far sentinels.
//   Phase B: pairwise d2 via V_WMMA_F32_16X16X4_F32 Gram tiles,
//            per-row min (NN), uniformity loss partial.
// ---------------------------------------------------------------------------
__global__ __launch_bounds__(BQ_THREADS)
void uniform_group_kernel(const float*  __restrict__ pts,
                          const float4* __restrict__ subpts,
                          float*        __restrict__ part,
                          int B)
{
    const int blk = blockIdx.x;
    const int pi  = blk / (B * NPOINT);
    const int rem = blk % (B * NPOINT);
    const int b   = rem / NPOINT;
    const int s   = rem % NPOINT;

    const float p       = c_p[pi];
    const int   nsample = c_ns[pi];
    const int   NT      = (nsample + 15) >> 4;   // 16-row tiles
    const int   NS16    = NT << 4;
    const float rr      = sqrtf(p);
    const float r2      = rr * rr;
    const float expect  = sqrtf(PI_F * p / (float)nsample);

    const int tid = threadIdx.x;
    const float* bp = pts + (size_t)b * N_PTS * 3;

    __shared__ float4 s_g[NS16_MAX];     // (x,y,z,|g|^2)
    __shared__ int    s_scan[BQ_THREADS];
    __shared__ float  s_wpart[16];

    const float4 C = subpts[b * NPOINT + s];

    // ---- Phase A: predicate + prefix scan + ordered compaction ----
    unsigned mask = 0;
    int cnt = 0;
#pragma unroll
    for (int k = 0; k < PPT; ++k) {
        const int pidx = tid * PPT + k;
        const float dx = bp[pidx * 3 + 0] - C.x;
        const float dy = bp[pidx * 3 + 1] - C.y;
        const float dz = bp[pidx * 3 + 2] - C.z;
        const float d2 = dx * dx + dy * dy + dz * dz;
        if (d2 <= r2) { mask |= 1u << k; ++cnt; }
    }
    s_scan[tid] = cnt;
    __syncthreads();
    for (int off = 1; off < BQ_THREADS; off <<= 1) {     // Hillis-Steele inclusive scan
        const int add = (tid >= off) ? s_scan[tid - off] : 0;
        __syncthreads();
        s_scan[tid] += add;
        __syncthreads();
    }
    const int base = s_scan[tid] - cnt;   // exclusive prefix
    const int T    = s_scan[BQ_THREADS - 1];

    int rank = base;
#pragma unroll
    for (int k = 0; k < PPT; ++k) {
        if (mask & (1u << k)) {
            if (rank < nsample) {
                const int pidx = tid * PPT + k;
                const float x = bp[pidx * 3 + 0];
                const float y = bp[pidx * 3 + 1];
                const float z = bp[pidx * 3 + 2];
                s_g[rank] = make_float4(x, y, z, x * x + y * y + z * z);
            }
            ++rank;
        }
    }
    __syncthreads();

    // pad: duplicate first point up to nsample (matches reference padding),
    // far sentinels up to the 16-multiple (never win the min)
    const int used = min(T, nsample);     // >= 1: the center is its own neighbor
    const float4 g0 = s_g[0];
    for (int q = used + tid; q < NS16; q += BQ_THREADS) {
        s_g[q] = (q < nsample) ? g0 : make_float4(1.0e6f, 1.0e6f, 1.0e6f, 3.0e12f);
    }
    __syncthreads();

    // ---- Phase B: WMMA Gram + row-min NN distance ----
    const int lane = tid & 31;
    const int wave = tid >> 5;
    const int rlo  = lane & 15;
    const int hi   = lane >> 4;

    float acc = 0.0f;
    for (int itile = wave; itile < NT; itile += 8) {
        const int i0 = itile << 4;
        // A 16x4 layout: lanes 0-15 hold (K0,K1)=(x,y) of row i0+lane,
        //                lanes 16-31 hold (K2,K3)=(z,0) of row i0+lane-16
        const float4 GA = s_g[i0 + rlo];
        v2f a;
        a.x = hi ? GA.z : GA.x;
        a.y = hi ? 0.0f : GA.y;

        float sqi[8], rowmin[8];
#pragma unroll
        for (int k = 0; k < 8; ++k) {
            sqi[k]    = s_g[i0 + k + 8 * hi].w;
            rowmin[k] = 3.0e38f;
        }

        for (int jtile = 0; jtile < NT; ++jtile) {
            const int j0 = jtile << 4;
            const float4 GB = s_g[j0 + rlo];
            v2f bb;                       // B 4x16 mirrors A's striping
            bb.x = hi ? GB.z : GB.x;
            bb.y = hi ? 0.0f : GB.y;
            const float sqj = GB.w;       // column j = j0 + rlo for both halves

            v8f cc = {};
            // D = A x B (+0): 16x16 tile of dot products
            v8f d = __builtin_amdgcn_wmma_f32_16x16x4_f32(
                false, a, false, bb, (short)0, cc, false, false);

            const int j = j0 + rlo;
#pragma unroll
            for (int k = 0; k < 8; ++k) {
                const int i = i0 + k + 8 * hi;     // C/D layout row
                float d2 = fmaxf(sqi[k] + sqj - 2.0f * d[k], 0.0f);
                d2 = (i == j) ? 1.0e10f : d2;      // mask diagonal
                rowmin[k] = fminf(rowmin[k], d2);
            }
        }

        // reduce min over the 16 columns held across each 16-lane half
#pragma unroll
        for (int k = 0; k < 8; ++k) {
            float v = rowmin[k];
            for (int m = 1; m < 16; m <<= 1)
                v = fminf(v, __shfl_xor(v, m, 32));
            const int row = i0 + k + 8 * hi;
            const float nn   = sqrtf(fmaxf(v, 1.0e-24f));
            const float diff = nn - expect;
            acc += (row < nsample) ? diff * diff : 0.0f;   // branchless, EXEC stays full
        }
    }

    if (rlo == 0) s_wpart[(wave << 1) | hi] = acc;   // waves with no tile write 0
    __syncthreads();
    if (tid == 0) {
        float total = 0.0f;
#pragma unroll
        for (int q = 0; q < 16; ++q) total += s_wpart[q];
        const float invE = 1.0f / (expect + 1.0e-12f);
        const float w100 = p * 100.0f;
        const float wgt  = (w100 * w100) /
                           ((float)NPOINT * (float)nsample * (float)NPERC);
        part[blk] = total * invE * wgt;
    }
}

// ---------------------------------------------------------------------------
// Kernel 3: deterministic fixed-order reduction of partials into loss[B]
// ---------------------------------------------------------------------------
__global__ void reduce_kernel(const float* __restrict__ part,
                              float* __restrict__ out, int B)
{
    const int b = threadIdx.x;
    if (b < B) {
        float acc = 0.0f;
        for (int pi = 0; pi < NPERC; ++pi)
            for (int s = 0; s < NPOINT; ++s)
                acc += part[(pi * B + b) * NPOINT + s];
        out[b] = acc;
    }
}

// ---------------------------------------------------------------------------
extern "C" void kernel_launch(void* const* d_in, const int* in_sizes, int n_in,
                              void* d_out, int out_size, void* d_ws, size_t ws_size,
                              hipStream_t stream)
{
    const float* pts = (const float*)d_in[0];
    const int B = in_sizes[0] / (N_PTS * 3);   // = 4 for the reference setup

    // workspace layout
    float4* subpts = (float4*)d_ws;
    const size_t sub_bytes = ((size_t)B * NPOINT * sizeof(float4) + 255) & ~(size_t)255;
    float* part = (float*)((char*)d_ws + sub_bytes);

    fps_kernel<<<B, BQ_THREADS, 0, stream>>>(pts, subpts);
    uniform_group_kernel<<<NPERC * B * NPOINT, BQ_THREADS, 0, stream>>>(pts, subpts, part, B);
    reduce_kernel<<<1, 32, 0, stream>>>(part, (float*)d_out, B);
}